// AttnDecoderRNN_25580825215617
// MI455X (gfx1250) — compile-verified
//
#include <hip/hip_runtime.h>
#include <math.h>

typedef __attribute__((ext_vector_type(2))) float v2f;
typedef __attribute__((ext_vector_type(8))) float v8f;

// global (address_space(1)) views — force global_load/global_store selection
typedef __attribute__((address_space(1))) const float  GCF;
typedef __attribute__((address_space(1)))       float  GF;
typedef __attribute__((address_space(1))) const v2f    GCV2F;

#define Bsz 64
#define Lsz 50
#define Hsz 1024
#define Vsz 50257

// ---------------------------------------------------------------------------
// Generic M=64 fp32 WMMA GEMM:  C[64,N] (=|+)= A[64,K] * B[K,N] (+bias)
// Block = 128 threads (4 waves). Wave w computes rows [16w,16w+16).
// Each wave owns a 16x64 output stripe: 4 independent 16x16 accumulators.
// K stepped by 4 using V_WMMA_F32_16X16X4_F32.
// A-frag layout (16x4 f32): lane holds m=lane&15, k = vgpr + 2*(lane>>4).
// B-frag layout (4x16 f32): lane holds n=lane&15, k = vgpr + 2*(lane>>4).
// C/D layout: vgpr g -> row = 16w + g + 8*(lane>>4), col = lane&15.
//
// Addressing: every stream is (uniform SGPR base) + (32-bit element voffset,
// scale_offset form). The NN fragment's second k-row uses a second uniform
// base (Bg2 = Bg + ldb) so it shares the SAME voffset register -> the whole
// inner loop has no 64-bit VALU address math. B loads are non-temporal
// (single-use weight streams; preserve L2 for enc/logits). Software-pipelined.
// ---------------------------------------------------------------------------
template<bool TRANS_B, bool ACCUM>
__global__ __launch_bounds__(128)
void gemm64_wmma(const float* __restrict__ A, int lda,
                 const float* __restrict__ B, int ldb,
                 const float* __restrict__ bias,
                 float* __restrict__ C, int ldc,
                 int N, int K)
{
  GCF*   __restrict__ Bg  = (GCF*)B;
  GCF*   __restrict__ Bg2 = (GCF*)B + (TRANS_B ? 0 : ldb); // 2nd k-row base
  GCV2F* __restrict__ Av2 = (GCV2F*)A;
  GCV2F* __restrict__ Bv2 = (GCV2F*)B;                     // TRANS_B stream
  GF*    __restrict__ Cg  = (GF*)C;

  const int tid  = threadIdx.x;
  const int wave = tid >> 5;
  const int lane = tid & 31;
  const int hi   = lane >> 4;
  const int lo   = lane & 15;
  const int m    = wave * 16 + lo;   // A row this lane fetches
  const int koff = hi * 2;           // k offset within a 4-wide step
  const int n0   = blockIdx.x * 64;

  int col[4], colc[4];
  v8f acc[4];
#pragma unroll
  for (int j = 0; j < 4; ++j) {
    col[j]  = n0 + j * 16 + lo;
    colc[j] = (col[j] < N) ? col[j] : (N - 1);   // clamp: EXEC stays all-1s
    if (ACCUM) {
#pragma unroll
      for (int g = 0; g < 8; ++g)
        acc[j][g] = Cg[(size_t)(wave * 16 + g + 8 * hi) * ldc + colc[j]];
    } else {
#pragma unroll
      for (int g = 0; g < 8; ++g) acc[j][g] = 0.0f;
    }
  }

  // --- 32-bit element-offset streams (constant stride per k-step) ---
  unsigned av2 = (unsigned)(m * lda + koff) >> 1;   // in v2f units (+2/step)
  unsigned boff[4];                                 // NN: float units
  unsigned bo2[4];                                  // TRANS: v2f units
#pragma unroll
  for (int j = 0; j < 4; ++j) {
    boff[j] = (unsigned)(koff * ldb + colc[j]);
    bo2[j]  = (unsigned)(colc[j] * ldb + koff) >> 1;
  }
  const unsigned bstep = 4u * (unsigned)ldb;        // NN float units / step

  // prefetch pointer: 32 lanes cover the full 4-row x 64-col B slab
  // (row = k+PF+(lane>>3), col = n0+(lane&7)*8) -> every 32B of each row hit
  const int PF = 32;  // rows (k) ahead
  const float* pfp = B + (size_t)(PF + (lane >> 3)) * ldb
                       + (size_t)n0 + (size_t)(lane & 7) * 8;

  // --- software-pipelined main loop ---
  v2f a, bf[4];
  a = Av2[av2];
#pragma unroll
  for (int j = 0; j < 4; ++j) {
    if (TRANS_B) {
      bf[j] = __builtin_nontemporal_load(Bv2 + bo2[j]);
    } else {
      bf[j][0] = __builtin_nontemporal_load(Bg  + boff[j]);
      bf[j][1] = __builtin_nontemporal_load(Bg2 + boff[j]);
    }
  }

  for (int k0 = 0; k0 < K - 4; k0 += 4) {
    // fetch next fragments first (hide latency behind the 4 WMMAs below)
    av2 += 2;
    v2f an = Av2[av2];
    v2f bn[4];
#pragma unroll
    for (int j = 0; j < 4; ++j) {
      if (TRANS_B) {
        bo2[j] += 2;
        bn[j] = __builtin_nontemporal_load(Bv2 + bo2[j]);
      } else {
        boff[j] += bstep;
        bn[j][0] = __builtin_nontemporal_load(Bg  + boff[j]);
        bn[j][1] = __builtin_nontemporal_load(Bg2 + boff[j]);
      }
    }
    if (!TRANS_B && (k0 + PF < K)) {        // uniform scalar branch
      __builtin_prefetch(pfp, 0, 1);
      pfp += (size_t)4 * ldb;
    }
#pragma unroll
    for (int j = 0; j < 4; ++j)
      acc[j] = __builtin_amdgcn_wmma_f32_16x16x4_f32(
          false, a, false, bf[j], (short)0, acc[j], false, false);
    a = an;
#pragma unroll
    for (int j = 0; j < 4; ++j) bf[j] = bn[j];
  }
#pragma unroll
  for (int j = 0; j < 4; ++j)
    acc[j] = __builtin_amdgcn_wmma_f32_16x16x4_f32(
        false, a, false, bf[j], (short)0, acc[j], false, false);

#pragma unroll
  for (int j = 0; j < 4; ++j) {
    if (col[j] < N) {
      const float bb = bias ? ((GCF*)bias)[col[j]] : 0.0f;
#pragma unroll
      for (int g = 0; g < 8; ++g)
        Cg[(size_t)(wave * 16 + g + 8 * hi) * ldc + col[j]] = acc[j][g] + bb;
    }
  }
}

// ---------------------------------------------------------------------------
// scores[b,l] = (qk[b,:]·enc[b,l,:] + q[b,:]·key_b) / H ; softmax over l ;
// u[b,:] = sum_l attn[b,l] * enc[b,l,:]
// One block per batch row. qk row staged in LDS (8 KB).
// ---------------------------------------------------------------------------
__global__ __launch_bounds__(256)
void attn_kernel(const float* __restrict__ q,      // [64,1024]
                 const float* __restrict__ qk,     // [64,2048]
                 const float* __restrict__ enc,    // [64,50,2048]
                 const float* __restrict__ key_b,  // [1024]
                 float* __restrict__ u,            // [64,2048]
                 float* __restrict__ attn_out)     // [64,50]
{
  __shared__ float qks[2 * Hsz];
  __shared__ float sc[64];
  __shared__ float red[256];
  const int b = blockIdx.x;
  const int tid = threadIdx.x;

  for (int i = tid; i < 2 * Hsz; i += 256) qks[i] = qk[(size_t)b * 2 * Hsz + i];

  float p = 0.0f;
  for (int i = tid; i < Hsz; i += 256) p += q[(size_t)b * Hsz + i] * key_b[i];
  red[tid] = p;
  __syncthreads();
  for (int s = 128; s > 0; s >>= 1) { if (tid < s) red[tid] += red[tid + s]; __syncthreads(); }
  const float qb = red[0];

  const int wave = tid >> 5, lane = tid & 31;
  for (int l = wave; l < Lsz; l += 8) {
    const float* er = enc + (size_t)b * Lsz * 2 * Hsz + (size_t)l * 2 * Hsz;
    float d = 0.0f;
    for (int c = lane; c < 2 * Hsz; c += 32) d += er[c] * qks[c];
    for (int off = 16; off > 0; off >>= 1) d += __shfl_xor(d, off, 32);
    if (lane == 0) sc[l] = (d + qb) * (1.0f / (float)Hsz);
  }
  __syncthreads();

  if (tid == 0) {
    float mx = -3.4e38f;
    for (int l = 0; l < Lsz; ++l) mx = fmaxf(mx, sc[l]);
    float s = 0.0f;
    for (int l = 0; l < Lsz; ++l) { float e = __expf(sc[l] - mx); sc[l] = e; s += e; }
    const float inv = 1.0f / s;
    for (int l = 0; l < Lsz; ++l) sc[l] *= inv;
  }
  __syncthreads();

  for (int l = tid; l < Lsz; l += 256) attn_out[(size_t)b * Lsz + l] = sc[l];

  for (int c = tid; c < 2 * Hsz; c += 256) {
    const float* ec = enc + (size_t)b * Lsz * 2 * Hsz + c;
    float acc = 0.0f;
    for (int l = 0; l < Lsz; ++l) acc += sc[l] * ec[(size_t)l * 2 * Hsz];
    u[(size_t)b * 2 * Hsz + c] = acc;
  }
}

__global__ __launch_bounds__(256)
void gather_emb(const int* __restrict__ inp, const float* __restrict__ emb,
                float* __restrict__ e)
{
  const int idx = blockIdx.x * 256 + threadIdx.x;   // 0 .. 64*1024-1
  const int b = idx >> 10, h = idx & 1023;
  e[idx] = emb[(size_t)inp[b] * Hsz + h];
}

__global__ __launch_bounds__(256)
void lstm_kernel(const float* __restrict__ gates,   // [64,4096] (f,i,c,o)
                 const float* __restrict__ c_prev,  // [64,1024]
                 float* __restrict__ h_out)         // [64,1024]
{
  const int idx = blockIdx.x * 256 + threadIdx.x;
  const int b = idx >> 10, h = idx & 1023;
  const float* g = gates + (size_t)b * 4 * Hsz;
  const float f  = 1.0f / (1.0f + __expf(-g[h]));
  const float i  = 1.0f / (1.0f + __expf(-g[Hsz + h]));
  const float cb = tanhf(g[2 * Hsz + h]);
  const float o  = 1.0f / (1.0f + __expf(-g[3 * Hsz + h]));
  const float cn = f * c_prev[idx] + i * cb;
  h_out[idx] = o * tanhf(cn);
}

__global__ __launch_bounds__(256)
void logsoftmax_kernel(const float* __restrict__ logits, float* __restrict__ out)
{
  __shared__ float red[256];
  const int row = blockIdx.x, tid = threadIdx.x;
  const float* x = logits + (size_t)row * Vsz;

  float mx = -3.4e38f;
  for (int i = tid; i < Vsz; i += 256) mx = fmaxf(mx, x[i]);
  red[tid] = mx;
  __syncthreads();
  for (int s = 128; s > 0; s >>= 1) { if (tid < s) red[tid] = fmaxf(red[tid], red[tid + s]); __syncthreads(); }
  mx = red[0];
  __syncthreads();

  float sum = 0.0f;
  for (int i = tid; i < Vsz; i += 256) sum += __expf(x[i] - mx);
  red[tid] = sum;
  __syncthreads();
  for (int s = 128; s > 0; s >>= 1) { if (tid < s) red[tid] += red[tid + s]; __syncthreads(); }
  const float lse = mx + __logf(red[0]);

  for (int i = tid; i < Vsz; i += 256) out[(size_t)row * Vsz + i] = x[i] - lse;
}

extern "C" void kernel_launch(void* const* d_in, const int* in_sizes, int n_in,
                              void* d_out, int out_size, void* d_ws, size_t ws_size,
                              hipStream_t stream)
{
  (void)in_sizes; (void)n_in; (void)out_size; (void)ws_size;

  const int*   inp     = (const int*)  d_in[0];
  const float* h_prev  = (const float*)d_in[1];
  const float* enc     = (const float*)d_in[2];
  const float* emb     = (const float*)d_in[3];
  const float* w_x     = (const float*)d_in[4];
  const float* b_x     = (const float*)d_in[5];
  const float* w_h     = (const float*)d_in[6];
  const float* b_h     = (const float*)d_in[7];
  const float* key_w   = (const float*)d_in[8];
  const float* key_b   = (const float*)d_in[9];
  const float* value_w = (const float*)d_in[10];
  const float* value_b = (const float*)d_in[11];
  const float* query_w = (const float*)d_in[12];
  const float* query_b = (const float*)d_in[13];
  const float* out_w   = (const float*)d_in[14];
  const float* out_b   = (const float*)d_in[15];

  float* out      = (float*)d_out;
  float* out_logp = out;                               // [64,50257]
  float* out_h    = out + (size_t)Bsz * Vsz;           // [64,1024]
  float* out_attn = out_h + (size_t)Bsz * Hsz;         // [64,50]

  float* ws = (float*)d_ws;
  size_t off = 0;
  float* q      = ws + off; off += (size_t)Bsz * Hsz;       // 64x1024
  float* qk     = ws + off; off += (size_t)Bsz * 2 * Hsz;   // 64x2048
  float* u      = ws + off; off += (size_t)Bsz * 2 * Hsz;   // 64x2048
  float* cprev  = ws + off; off += (size_t)Bsz * Hsz;       // 64x1024
  float* gates  = ws + off; off += (size_t)Bsz * 4 * Hsz;   // 64x4096
  float* e      = ws + off; off += (size_t)Bsz * Hsz;       // 64x1024
  float* logits = ws + off;                                 // 64x50257

  const dim3 gblk(128);

  // 1. q = h_prev @ query_w + query_b            [64,1024] K=1024
  gemm64_wmma<false,false><<<dim3(Hsz / 64), gblk, 0, stream>>>(
      h_prev, Hsz, query_w, Hsz, query_b, q, Hsz, Hsz, Hsz);

  // 2. qk = q @ key_w^T                          [64,2048] K=1024
  gemm64_wmma<true,false><<<dim3(2 * Hsz / 64), gblk, 0, stream>>>(
      q, Hsz, key_w, Hsz, nullptr, qk, 2 * Hsz, 2 * Hsz, Hsz);

  // 3. scores -> softmax -> attn (output) ; u = attn @ enc
  attn_kernel<<<dim3(Bsz), dim3(256), 0, stream>>>(q, qk, enc, key_b, u, out_attn);

  // 4. c_prev = u @ value_w + value_b            [64,1024] K=2048
  gemm64_wmma<false,false><<<dim3(Hsz / 64), gblk, 0, stream>>>(
      u, 2 * Hsz, value_w, Hsz, value_b, cprev, Hsz, Hsz, 2 * Hsz);

  // 5. embedded = emb[input]
  gather_emb<<<dim3(Bsz * Hsz / 256), dim3(256), 0, stream>>>(inp, emb, e);

  // 6. gates = e @ w_x + b_x ; gates += h_prev @ w_h + b_h
  gemm64_wmma<false,false><<<dim3(4 * Hsz / 64), gblk, 0, stream>>>(
      e, Hsz, w_x, 4 * Hsz, b_x, gates, 4 * Hsz, 4 * Hsz, Hsz);
  gemm64_wmma<false,true><<<dim3(4 * Hsz / 64), gblk, 0, stream>>>(
      h_prev, Hsz, w_h, 4 * Hsz, b_h, gates, 4 * Hsz, 4 * Hsz, Hsz);

  // 7. LSTM elementwise -> h_next (output region, reused as GEMM A)
  lstm_kernel<<<dim3(Bsz * Hsz / 256), dim3(256), 0, stream>>>(gates, cprev, out_h);

  // 8. logits = h_next @ out_w + out_b           [64,50257] K=1024
  gemm64_wmma<false,false><<<dim3((Vsz + 63) / 64), gblk, 0, stream>>>(
      out_h, Hsz, out_w, Vsz, out_b, logits, Vsz, Vsz, Hsz);

  // 9. log_softmax rows -> output
  logsoftmax_kernel<<<dim3(Bsz), dim3(256), 0, stream>>>(logits, out_logp);
}